// Yololayer_13065290514748
// MI455X (gfx1250) — compile-verified
//
#include <hip/hip_runtime.h>
#include <stdint.h>

// Problem geometry (fixed by the reference).
#define B_    64
#define A_    3
#define NC_   80
#define H_    80
#define W_    80
#define HW_   (H_ * W_)        // 6400
#define CH_   (5 + NC_)        // 85 channels per anchor
#define TPB   256              // 8 waves (wave32)
#define CHUNK 16               // class channels per async-staged chunk
#define NCHUNK (NC_ / CHUNK)   // 5
#define TILES  (HW_ / TPB)     // 25

__device__ __forceinline__ float sigmoidf_(float x) {
    return 1.0f / (1.0f + __expf(-x));
}

// boxes region: 7 * B*A*HW floats, then keep mask: B*A*HW floats (0/1).
extern "C" __global__ __launch_bounds__(TPB, 1)
void yolo_region_kernel(const float* __restrict__ in,      // [B, A*85, H, W]
                        const float* __restrict__ anchors, // [3,2] grid units
                        const float* __restrict__ nms,     // [1]
                        float* __restrict__ out)
{
    // Double-buffered staging: 2 bufs x 16 channels x 256 cells x 4B = 32 KB
    __shared__ __align__(16) float sm[2 * CHUNK * TPB];

    const int t    = threadIdx.x;
    const int tile = blockIdx.x % TILES;
    const int ba   = blockIdx.x / TILES;      // b*A + a (uniform -> SGPR)
    const int a    = ba % A_;
    const int hw   = tile * TPB + t;

    const float* base = in + (size_t)ba * CH_ * HW_;

    // ---- b128 async staging: each wave stages its own 32 cells ----------
    // One instruction: 32 lanes x 16B = 4 channels x 32 cells (512 B).
    // Lane l covers channel sub-index (l>>3) and cells (l&7)*4 .. +3.
    const int l     = t & 31;
    const int wv    = t >> 5;
    const int ccl   = l >> 3;                  // 0..3
    const int cellb = wv * 32 + (l & 7) * 4;   // first of 4 cells staged, 16B aligned
    const uint32_t lds0      = (uint32_t)(size_t)(&sm[0]);
    const uint32_t ldsa_base = lds0 + (uint32_t)((ccl * TPB + cellb) * 4);
    const uint32_t voff_base = (uint32_t)(((5 + ccl) * HW_ + tile * TPB + cellb) * 4);

    auto issue_chunk = [&](int chunk, int buf) {
        #pragma unroll
        for (int j = 0; j < 4; ++j) {          // 4 async instrs = 16 ch x 32 cells
            uint32_t ldsa = ldsa_base + (uint32_t)((buf * CHUNK + 4 * j) * TPB * 4);
            uint32_t voff = voff_base + (uint32_t)((chunk * CHUNK + 4 * j) * HW_ * 4);
            asm volatile("global_load_async_to_lds_b128 %0, %1, %2"
                         :: "v"(ldsa), "v"(voff), "s"(base)
                         : "memory");
        }
    };

    // Prime the pipeline: two chunks (4 async ops each, per wave) in flight.
    issue_chunk(0, 0);
    issue_chunk(1, 1);

    // Overlap: coalesced, non-temporal loads of the 5 box channels.
    const float tx = __builtin_nontemporal_load(base + 0 * HW_ + hw);
    const float ty = __builtin_nontemporal_load(base + 1 * HW_ + hw);
    const float tw = __builtin_nontemporal_load(base + 2 * HW_ + hw);
    const float th = __builtin_nontemporal_load(base + 3 * HW_ + hw);
    const float tc = __builtin_nontemporal_load(base + 4 * HW_ + hw);

    // Softmax-max over 80 classes: max(softmax) = 1 / sum(exp(v - max)).
    float m  = -3.4028235e38f;   // running global max
    float s  = 0.0f;             // running sum of exp(v - m)
    int   id = 0;                // argmax

    #pragma unroll
    for (int k = 0; k < NCHUNK; ++k) {
        // Async loads complete in order: <=4 outstanding => chunk k landed.
        if (k < NCHUNK - 1) {
            asm volatile("s_wait_asynccnt 4" ::: "memory");
        } else {
            asm volatile("s_wait_asynccnt 0" ::: "memory");
        }

        const int buf = k & 1;

        // Pull the staged chunk into registers (one DS read per element).
        float r[CHUNK];
        #pragma unroll
        for (int cc = 0; cc < CHUNK; ++cc)
            r[cc] = sm[(buf * CHUNK + cc) * TPB + t];

        // Refill this buffer once our LDS reads have retired.
        if (k + 2 < NCHUNK) {
            asm volatile("s_wait_dscnt 0" ::: "memory");
            issue_chunk(k + 2, buf);
        }

        // Pass 1: running max + argmax (strict '>' keeps first index on ties).
        const float m_old = m;
        #pragma unroll
        for (int cc = 0; cc < CHUNK; ++cc) {
            const int c = k * CHUNK + cc;
            id = (r[cc] > m) ? c : id;
            m  = fmaxf(m, r[cc]);
        }

        // Pass 2: one exp per element, 4-way accumulators, then merge.
        float s0 = 0.0f, s1 = 0.0f, s2 = 0.0f, s3 = 0.0f;
        #pragma unroll
        for (int cc = 0; cc < CHUNK; cc += 4) {
            s0 += __expf(r[cc + 0] - m);
            s1 += __expf(r[cc + 1] - m);
            s2 += __expf(r[cc + 2] - m);
            s3 += __expf(r[cc + 3] - m);
        }
        s = s * __expf(m_old - m) + ((s0 + s1) + (s2 + s3));
    }

    // Box decode.
    const int gy = hw / W_;
    const int gx = hw - gy * W_;
    const float invW = 1.0f / (float)W_;
    const float invH = 1.0f / (float)H_;

    const float bx   = (sigmoidf_(tx) + (float)gx) * invW;
    const float by   = (sigmoidf_(ty) + (float)gy) * invH;
    const float bw   = __expf(tw) * anchors[a * 2 + 0] * invW;
    const float bh   = __expf(th) * anchors[a * 2 + 1] * invH;
    const float det  = sigmoidf_(tc);
    const float cmax = 1.0f / s;   // max softmax prob

    const size_t cell = (size_t)ba * HW_ + (size_t)hw;
    float* bo = out + cell * 7;
    __builtin_nontemporal_store(bx,         bo + 0);
    __builtin_nontemporal_store(by,         bo + 1);
    __builtin_nontemporal_store(bw,         bo + 2);
    __builtin_nontemporal_store(bh,         bo + 3);
    __builtin_nontemporal_store(det,        bo + 4);
    __builtin_nontemporal_store(cmax,       bo + 5);
    __builtin_nontemporal_store((float)id,  bo + 6);

    // keep mask appended after the boxes region, as float 0/1.
    __builtin_nontemporal_store((det > nms[0]) ? 1.0f : 0.0f,
                                out + (size_t)7 * B_ * A_ * HW_ + cell);
}

extern "C" void kernel_launch(void* const* d_in, const int* in_sizes, int n_in,
                              void* d_out, int out_size, void* d_ws, size_t ws_size,
                              hipStream_t stream) {
    const float* in      = (const float*)d_in[0];
    const float* anchors = (const float*)d_in[1];
    const float* nms     = (const float*)d_in[2];
    float* out           = (float*)d_out;

    const int blocks = B_ * A_ * TILES;   // 64*3*25 = 4800
    hipLaunchKernelGGL(yolo_region_kernel, dim3(blocks), dim3(TPB), 0, stream,
                       in, anchors, nms, out);
}